// HSL_Layer_Part1_58944131170869
// MI455X (gfx1250) — compile-verified
//
#include <hip/hip_runtime.h>
#include <hip/hip_bf16.h>
#include <math.h>

#define N_NODES 1024
#define M_SEG   1024
#define DDIM    128
#define NNZ_E   32768
#define HDIM    256

typedef __attribute__((ext_vector_type(2))) float v2f;
typedef __attribute__((ext_vector_type(8))) float v8f;

// ---------------------------------------------------------------- zero ws ---
__global__ void k_zero(float* __restrict__ p, int n) {
    int i = blockIdx.x * blockDim.x + threadIdx.x;
    if (i < n) p[i] = 0.0f;
}

// ------------------------------------------------------------ scatter-sum ---
// 32 lanes per edge; each lane atomically adds 4 consecutive d-elements.
__global__ void k_scatter(const float* __restrict__ X,
                          const int*   __restrict__ V,
                          const int*   __restrict__ E,
                          float* __restrict__ sums,
                          float* __restrict__ counts) {
    int t    = blockIdx.x * blockDim.x + threadIdx.x;
    int j    = t >> 5;
    int lane = t & 31;
    if (j >= NNZ_E) return;
    int v = V[j], m = E[j];
    const float* xr = X    + (size_t)v * DDIM + lane * 4;
    float*       sr = sums + (size_t)m * DDIM + lane * 4;
    atomicAdd(sr + 0, xr[0]);
    atomicAdd(sr + 1, xr[1]);
    atomicAdd(sr + 2, xr[2]);
    atomicAdd(sr + 3, xr[3]);
    if (lane == 0) atomicAdd(counts + m, 1.0f);
}

// -------------------------------------------------- WMMA f32 16x16x4 GEMM ---
// One wave per 16x16 output tile. Tiles [0,1024): hx = X @ W1[:D].
// Tiles [1024,2048): he = (sums/max(cnt,1)) @ W1[D:] + b1  (div + bias fused).
__global__ __launch_bounds__(256)
void k_gemm_wmma(const float* __restrict__ X,
                 const float* __restrict__ sums,
                 const float* __restrict__ counts,
                 const float* __restrict__ W1,
                 const float* __restrict__ b1,
                 float* __restrict__ hxb,
                 float* __restrict__ heb) {
    const int lane  = threadIdx.x & 31;
    const int wave  = threadIdx.x >> 5;
    const int tile  = blockIdx.x * (blockDim.x >> 5) + wave;   // 0..2047
    const int tilesPerMat = (N_NODES / 16) * (HDIM / 16);      // 1024

    const bool isHE = (tile >= tilesPerMat);
    const int  t    = isHE ? tile - tilesPerMat : tile;
    const int  r0   = (t / (HDIM / 16)) * 16;   // row tile base (n or m)
    const int  c0   = (t % (HDIM / 16)) * 16;   // col tile base (h)

    const int mrow  = lane & 15;   // A-row / B-col / D-col owned by this lane
    const int khalf = lane >> 4;   // which K half-pair this lane holds

    // A-operand source row (with fused mean-division for the he GEMM)
    float inv = 1.0f;
    const float* Arow;
    if (isHE) {
        inv  = 1.0f / fmaxf(counts[r0 + mrow], 1.0f);
        Arow = sums + (size_t)(r0 + mrow) * DDIM;
    } else {
        Arow = X + (size_t)(r0 + mrow) * DDIM;
    }
    const float* Wb = W1 + (size_t)(isHE ? DDIM : 0) * HDIM;

    v8f acc = {};
    for (int k = 0; k < DDIM; k += 4) {
        v2f a, b;
        // A 16x4: lane holds A[mrow][k + 2*khalf + v]
        a.x = Arow[k + 2 * khalf + 0] * inv;
        a.y = Arow[k + 2 * khalf + 1] * inv;
        // B 4x16: VGPR v holds row K = k + v + 2*khalf, col = c0+mrow
        b.x = Wb[(size_t)(k + 0 + 2 * khalf) * HDIM + c0 + mrow];
        b.y = Wb[(size_t)(k + 1 + 2 * khalf) * HDIM + c0 + mrow];
        acc = __builtin_amdgcn_wmma_f32_16x16x4_f32(
            /*neg_a=*/false, a, /*neg_b=*/false, b,
            /*c_mod=*/(short)0, acc, /*reuse_a=*/false, /*reuse_b=*/false);
    }

    // D 16x16: VGPR v -> row r0 + v + 8*khalf, col c0 + mrow
    float* outp = isHE ? heb : hxb;
    const int col = c0 + mrow;
    const float badd = isHE ? b1[col] : 0.0f;
#pragma unroll
    for (int v = 0; v < 8; ++v) {
        int row = r0 + v + 8 * khalf;
        outp[(size_t)row * HDIM + col] = acc[v] + badd;
    }
}

// ------------------------------------------- pairwise relu-dot + sigmoid  ---
// out[n,m] = sigmoid( sum_h relu(hx[n,h] + he[m,h]) * W2[h] + b2 )
// 64x64 tile per block, 4x4 outputs per thread, H chunked by 32 through LDS.
#define TN 64
#define TM 64
#define HC 32

__global__ __launch_bounds__(256)
void k_pair(const float* __restrict__ hxb,
            const float* __restrict__ heb,
            const float* __restrict__ W2,
            const float* __restrict__ b2,
            float* __restrict__ out) {
    __shared__ float sx[TN][HC + 1];
    __shared__ float se[TM][HC + 1];
    __shared__ float sw[HDIM];

    const int tid = threadIdx.x;
    const int n0  = blockIdx.y * TN;
    const int m0  = blockIdx.x * TM;
    const int tx  = tid & 15;    // m-direction
    const int ty  = tid >> 4;    // n-direction

    sw[tid] = W2[tid];           // HDIM == blockDim == 256

    float acc[4][4] = {};
    for (int h0 = 0; h0 < HDIM; h0 += HC) {
        __syncthreads();
        for (int i = tid; i < TN * HC; i += 256) {
            int r = i >> 5, c = i & 31;
            sx[r][c] = hxb[(size_t)(n0 + r) * HDIM + h0 + c];
            se[r][c] = heb[(size_t)(m0 + r) * HDIM + h0 + c];
        }
        __syncthreads();
#pragma unroll 4
        for (int h = 0; h < HC; ++h) {
            const float w = sw[h0 + h];
            float a[4], b[4];
#pragma unroll
            for (int i = 0; i < 4; ++i) a[i] = sx[ty * 4 + i][h];
#pragma unroll
            for (int j = 0; j < 4; ++j) b[j] = se[tx * 4 + j][h];
#pragma unroll
            for (int i = 0; i < 4; ++i)
#pragma unroll
                for (int j = 0; j < 4; ++j)
                    acc[i][j] += fmaxf(a[i] + b[j], 0.0f) * w;
        }
    }

    const float bb = b2[0];
#pragma unroll
    for (int i = 0; i < 4; ++i) {
        const int n = n0 + ty * 4 + i;
#pragma unroll
        for (int j = 0; j < 4; ++j) {
            const int m = m0 + tx * 4 + j;
            const float x = acc[i][j] + bb;
            out[(size_t)n * M_SEG + m] = 1.0f / (1.0f + __expf(-x));
        }
    }
}

// ---------------------------------------------------------------- launch  ---
extern "C" void kernel_launch(void* const* d_in, const int* in_sizes, int n_in,
                              void* d_out, int out_size, void* d_ws, size_t ws_size,
                              hipStream_t stream) {
    (void)in_sizes; (void)n_in; (void)out_size; (void)ws_size;
    const float* X  = (const float*)d_in[0];
    const int*   V  = (const int*)  d_in[1];
    const int*   E  = (const int*)  d_in[2];
    const float* W1 = (const float*)d_in[3];
    const float* b1 = (const float*)d_in[4];
    const float* W2 = (const float*)d_in[5];
    const float* b2 = (const float*)d_in[6];
    float* out = (float*)d_out;

    // workspace layout (floats): sums[M*D] | counts[M] | hx[N*H] | he[M*H]
    float* sums   = (float*)d_ws;
    float* counts = sums + (size_t)M_SEG * DDIM;
    float* hxb    = counts + M_SEG;
    float* heb    = hxb + (size_t)N_NODES * HDIM;

    const int zn = M_SEG * DDIM + M_SEG;
    k_zero<<<(zn + 255) / 256, 256, 0, stream>>>(sums, zn);

    k_scatter<<<(NNZ_E * 32) / 256, 256, 0, stream>>>(X, V, E, sums, counts);

    // 2048 wave-tiles (hx: 1024, he: 1024), 8 waves / block
    k_gemm_wmma<<<256, 256, 0, stream>>>(X, sums, counts, W1, b1, hxb, heb);

    dim3 g3(M_SEG / TM, N_NODES / TN);
    k_pair<<<g3, 256, 0, stream>>>(hxb, heb, W2, b2, out);
}